// RejectionSampler_44040594653445
// MI455X (gfx1250) — compile-verified
//
#include <hip/hip_runtime.h>
#include <stdint.h>

// Problem constants (from reference: B=64, N=8, V=128000)
#define B_        64
#define N_        8
#define V_        128000
#define CHUNK     1024                  // floats per streamed chunk (4 KB)
#define NCHUNK    (V_ / CHUNK)          // 125, exact
#define ROWS_RES  (B_ * N_)             // 512 residual rows
#define ROWS_TOT  (ROWS_RES + B_)       // + 64 bonus rows = 576

typedef unsigned int u32x4 __attribute__((ext_vector_type(4)));
typedef int          i32x4 __attribute__((ext_vector_type(4)));
typedef int          i32x8 __attribute__((ext_vector_type(8)));
typedef float        f32x4 __attribute__((ext_vector_type(4)));

// ---------------- stateless RNG (deterministic across graph replays) -------
__device__ __forceinline__ uint32_t hash_u32(uint32_t x) {
  x ^= x >> 17; x *= 0xed5ad4bbu;
  x ^= x >> 11; x *= 0xac4c1b51u;
  x ^= x >> 15; x *= 0x31848babu;
  x ^= x >> 14;
  return x;
}
__device__ __forceinline__ float u_open01(uint32_t h) {
  // (0, 1]: never 0 so log() is finite
  return ((h >> 8) + 1u) * (1.0f / 16777216.0f);
}

// ---------------- Tensor Data Mover: 1-D tile Global -> LDS ---------------
// Unconditional: 5-arg form passed the frontend on this toolchain (ROCm 7.2).
__device__ __forceinline__ void tdm_load_chunk(const float* gsrc,
                                               uint32_t lds_off,
                                               int nelem) {
  uint64_t ga = (uint64_t)(uintptr_t)gsrc;
  // D# group 0 (ISA 08 §8.3): count=1 valid, lds_addr[63:32],
  // global_addr[120:64], type=2 ("image") in [127:126]
  u32x4 g0;
  g0[0] = 1u;
  g0[1] = lds_off;
  g0[2] = (uint32_t)ga;
  g0[3] = (uint32_t)((ga >> 32) & 0x01FFFFFFull) | 0x80000000u;
  // D# group 1 (§8.4): workgroup_mask=0 (not in cluster), data_size=2 (4B),
  // tensor_dim0 = nelem, tensor_dim1 = 1, tile_dim0 = nelem, tile_dim1 = 1,
  // tensor_dim0_stride = nelem. 1-D tile: groups 2/3 zero.
  i32x8 g1;
  g1[0] = (2 << 16);
  g1[1] = (nelem & 0xFFFF) << 16;
  g1[2] = ((nelem >> 16) & 0xFFFF) | (1 << 16);
  g1[3] = (nelem & 0xFFFF) << 16;
  g1[4] = 1;
  g1[5] = nelem;
  g1[6] = 0;
  g1[7] = 0;
  i32x4 z4 = {0, 0, 0, 0};
#if defined(__clang_major__) && (__clang_major__ >= 23)
  i32x8 z8 = {0, 0, 0, 0, 0, 0, 0, 0};
  __builtin_amdgcn_tensor_load_to_lds(g0, g1, z4, z4, z8, 0);
#else
  __builtin_amdgcn_tensor_load_to_lds(g0, g1, z4, z4, 0);
#endif
}

__device__ __forceinline__ void wait_tensorcnt_le1() {
  __builtin_amdgcn_s_wait_tensorcnt(1);
}
__device__ __forceinline__ void wait_tensorcnt_le0() {
  __builtin_amdgcn_s_wait_tensorcnt(0);
}

// ---------------------------------------------------------------------------
// Kernel 1: one workgroup per row (512 residual rows + 64 bonus rows).
// Streams verify_probs chunk via TDM into double-buffered LDS; draft_probs
// chunk straight to VGPRs (non-temporal b128) with prefetch of the next
// chunk. Fused residual + exponential-race categorical sampling:
//   sample ~ Categorical(r / sum r)  ==  argmax_v  r_v / Exp(1)_v
// (scale invariant, so the normalization pass of the reference is not needed)
// ---------------------------------------------------------------------------
__global__ __launch_bounds__(256)
void rs_row_sample_kernel(const float* __restrict__ draft_probs,
                          const float* __restrict__ verify_probs,
                          int* __restrict__ rec_out /* ROWS_TOT ints */) {
  __shared__ __align__(16) float lbuf[2][CHUNK];   // 8 KB double buffer
  __shared__ float red_k[8];
  __shared__ int   red_i[8];

  const int  row    = blockIdx.x;
  const bool is_res = (row < ROWS_RES);
  const int  b      = is_res ? (row >> 3) : (row - ROWS_RES);
  const int  n      = is_res ? (row & 7) : N_;

  const float* vrow = verify_probs + ((size_t)b * (N_ + 1) + n) * V_;
  const float* drow = is_res ? (draft_probs + ((size_t)b * N_ + n) * V_)
                             : (const float*)nullptr;

  const int tid  = threadIdx.x;
  const int wave = tid >> 5;
  const int lane = tid & 31;
  const uint32_t seed = 0x9E3779B9u ^ ((uint32_t)row * 0x85EBCA6Bu);

  const uint32_t lds0 = (uint32_t)(uintptr_t)(&lbuf[0][0]);
  const uint32_t lds1 = (uint32_t)(uintptr_t)(&lbuf[1][0]);

  if (wave == 0) {                      // wave-uniform; TDM ignores EXEC
    tdm_load_chunk(vrow + 0 * CHUNK, lds0, CHUNK);
    tdm_load_chunk(vrow + 1 * CHUNK, lds1, CHUNK);
  }

  float bestK = -__builtin_inff();
  int   bestI = 0;

  for (int c = 0; c < NCHUNK; ++c) {
    const int buf = c & 1;
    if (wave == 0) {
      if (c + 2 < NCHUNK) wait_tensorcnt_le1();   // chunk c done (in-order)
      else                wait_tensorcnt_le0();
    }
    __syncthreads();

    if (is_res && (c + 1 < NCHUNK))
      __builtin_prefetch(drow + (c + 1) * CHUNK + tid * 4, 0, 1);

    const f32x4 tv = reinterpret_cast<const f32x4*>(&lbuf[buf][0])[tid];
    f32x4 td = {0.f, 0.f, 0.f, 0.f};
    if (is_res)   // streamed once, 557 MB total >> L2: keep it non-temporal
      td = __builtin_nontemporal_load(
               reinterpret_cast<const f32x4*>(drow + c * CHUNK) + tid);

    const int base = c * CHUNK + tid * 4;
    const float rv[4] = {tv.x - td.x, tv.y - td.y, tv.z - td.z, tv.w - td.w};
#pragma unroll
    for (int j = 0; j < 4; ++j) {
      const float r = rv[j];
      if (r > 0.0f) {
        const uint32_t h = hash_u32(seed ^ (uint32_t)(base + j));
        const float e   = -__logf(u_open01(h));    // Exp(1) variate
        const float key = __fdividef(r, e);
        if (key > bestK) { bestK = key; bestI = base + j; }
      }
    }
    __syncthreads();
    if (wave == 0 && (c + 2 < NCHUNK))
      tdm_load_chunk(vrow + (c + 2) * CHUNK, buf ? lds1 : lds0, CHUNK);
  }

  // wave32 reduction, then cross-wave combine (tie -> lowest index)
#pragma unroll
  for (int off = 16; off > 0; off >>= 1) {
    const float ok = __shfl_xor(bestK, off, 32);
    const int   oi = __shfl_xor(bestI, off, 32);
    if (ok > bestK || (ok == bestK && oi < bestI)) { bestK = ok; bestI = oi; }
  }
  if (lane == 0) { red_k[wave] = bestK; red_i[wave] = bestI; }
  __syncthreads();
  if (tid == 0) {
    float bk = red_k[0]; int bi = red_i[0];
#pragma unroll
    for (int w = 1; w < 8; ++w)
      if (red_k[w] > bk || (red_k[w] == bk && red_i[w] < bi)) {
        bk = red_k[w]; bi = red_i[w];
      }
    rec_out[row] = bi;
  }
}

// ---------------------------------------------------------------------------
// Kernel 2: per-batch epilogue. Gathers p/q at the draft tokens, runs the
// accept chain, picks recovered/bonus token, emits all three outputs (float).
// ---------------------------------------------------------------------------
__global__ __launch_bounds__(64)
void rs_finalize_kernel(const int*   __restrict__ draft_ids,
                        const float* __restrict__ draft_probs,
                        const float* __restrict__ verify_probs,
                        const int*   __restrict__ rec,
                        float*       __restrict__ out) {
  const int b = threadIdx.x;
  if (b >= B_) return;

  int  acc = 0, em = 0;
  bool alive = true;
#pragma unroll
  for (int n = 0; n < N_; ++n) {
    const int   tok = draft_ids[b * N_ + n];
    const float p = verify_probs[((size_t)b * (N_ + 1) + n) * V_ + tok];
    const float q = draft_probs[((size_t)b * N_ + n) * V_ + tok];
    const float u =
        u_open01(hash_u32(0xA511E9B3u ^ ((uint32_t)(b * N_ + n) * 0x9E3779B9u)));
    const bool a = (u * q) < p;
    acc += a ? 1 : 0;
    if (alive && a) em += 1; else alive = false;
  }

  const int stop_n    = (em < N_ - 1) ? em : (N_ - 1);
  const int final_tok = (em < N_) ? rec[b * N_ + stop_n] : rec[ROWS_RES + b];

#pragma unroll
  for (int pos = 0; pos <= N_; ++pos) {
    float o;
    if (pos < em)       o = (float)draft_ids[b * N_ + pos];
    else if (pos == em) o = (float)final_tok;
    else                o = -1.0f;
    out[b * (N_ + 1) + pos] = o;
  }
  out[B_ * (N_ + 1) + b]      = (float)acc;   // accepted_token_num
  out[B_ * (N_ + 1) + B_ + b] = (float)em;    // emitted_token_num
}

// ---------------------------------------------------------------------------
extern "C" void kernel_launch(void* const* d_in, const int* in_sizes, int n_in,
                              void* d_out, int out_size, void* d_ws,
                              size_t ws_size, hipStream_t stream) {
  (void)in_sizes; (void)n_in; (void)out_size; (void)ws_size;
  const int*   draft_ids    = (const int*)d_in[0];
  const float* draft_probs  = (const float*)d_in[1];
  const float* verify_probs = (const float*)d_in[2];
  int* rec = (int*)d_ws;  // ROWS_TOT ints of scratch

  rs_row_sample_kernel<<<ROWS_TOT, 256, 0, stream>>>(draft_probs, verify_probs,
                                                     rec);
  rs_finalize_kernel<<<1, 64, 0, stream>>>(draft_ids, draft_probs, verify_probs,
                                           rec, (float*)d_out);
}